// GCRNN_time_19473381720838
// MI455X (gfx1250) — compile-verified
//
#include <hip/hip_runtime.h>

typedef __attribute__((ext_vector_type(2))) float v2f;
typedef __attribute__((ext_vector_type(8))) float v8f;
// explicit address spaces: LDS accesses -> ds_load/ds_store (32-bit addressing),
// weight/input accesses -> global_load (LOADcnt only, no flat/DScnt coupling)
typedef __attribute__((address_space(3))) float       lfloat;
typedef __attribute__((address_space(1))) const float gfloat;
typedef __attribute__((address_space(1))) float       gmfloat;

#define NB 128
#define NF 3
#define NN 64
#define NS 96
#define ND 64
#define ST 68          // LDS row stride in floats; 68 % 64 == 4 -> conflict-free column reads
#define NTH 512
#define NWV 16

__device__ __forceinline__ v8f wmma16(v2f a, v2f b, v8f c) {
  // V_WMMA_F32_16X16X4_F32 : D = A(16x4) * B(4x16) + C(16x16), fp32 throughout
  return __builtin_amdgcn_wmma_f32_16x16x4_f32(false, a, false, b, (short)0, c, false, false);
}

// One 16x16 C tile per wave; A in LDS, B in global memory.
// BT=true : B[k][c] = Bbase[c*bs + k]  (operand is W row-major, multiply by W^T)
// K must be a multiple of 16; loads are blocked 4 chunks ahead of 4 WMMAs.
template<bool BT>
__device__ __forceinline__ v8f tile_gemm_g(lfloat* Abase, int as,
                                           gfloat* Bbase, int bs,
                                           int K, v8f acc) {
  const int lane = threadIdx.x & 31;
  const int r    = lane & 15;
  const int ko   = (lane >> 4) << 1;       // 0 for lanes 0-15, 2 for lanes 16-31
#pragma unroll
  for (int k0 = 0; k0 < K; k0 += 16) {
    v2f a[4], b[4];
#pragma unroll
    for (int j = 0; j < 4; ++j) {
      const int k = k0 + 4 * j + ko;
      a[j].x = Abase[r * as + k];
      a[j].y = Abase[r * as + k + 1];
      if (BT) { b[j].x = Bbase[r * bs + k];       b[j].y = Bbase[r * bs + k + 1]; }
      else    { b[j].x = Bbase[k * bs + r];       b[j].y = Bbase[(k + 1) * bs + r]; }
    }
#pragma unroll
    for (int j = 0; j < 4; ++j) acc = wmma16(a[j], b[j], acc);
  }
  return acc;
}

// Same, but B also lives in LDS.
template<bool BT>
__device__ __forceinline__ v8f tile_gemm_l(lfloat* Abase, int as,
                                           lfloat* Bbase, int bs,
                                           int K, v8f acc) {
  const int lane = threadIdx.x & 31;
  const int r    = lane & 15;
  const int ko   = (lane >> 4) << 1;
#pragma unroll
  for (int k0 = 0; k0 < K; k0 += 16) {
    v2f a[4], b[4];
#pragma unroll
    for (int j = 0; j < 4; ++j) {
      const int k = k0 + 4 * j + ko;
      a[j].x = Abase[r * as + k];
      a[j].y = Abase[r * as + k + 1];
      if (BT) { b[j].x = Bbase[r * bs + k];       b[j].y = Bbase[r * bs + k + 1]; }
      else    { b[j].x = Bbase[k * bs + r];       b[j].y = Bbase[(k + 1) * bs + r]; }
    }
#pragma unroll
    for (int j = 0; j < 4; ++j) acc = wmma16(a[j], b[j], acc);
  }
  return acc;
}

// per-row layernorm over 64 cols; 8 threads per row (512 threads / 64 rows)
__device__ __forceinline__ void ln_rows(lfloat* src, lfloat* dst,
                                        gfloat* g, gfloat* be) {
  const int row = threadIdx.x >> 3, sub = threadIdx.x & 7;
  lfloat* rp = src + row * ST;
  float s = 0.f, s2 = 0.f;
  for (int j = sub; j < 64; j += 8) { float v = rp[j]; s += v; s2 += v * v; }
  for (int o = 1; o < 8; o <<= 1) { s += __shfl_xor(s, o); s2 += __shfl_xor(s2, o); }
  float mu  = s * (1.f / 64.f);
  float var = s2 * (1.f / 64.f) - mu * mu;
  float inv = rsqrtf(var + 1e-5f);
  lfloat* dp = dst + row * ST;
  for (int j = sub; j < 64; j += 8) dp[j] = (rp[j] - mu) * inv * g[j] + be[j];
}

__device__ void encoder_layer(
    lfloat* IN, lfloat* OUT,
    lfloat* Q, lfloat* Kb, lfloat* V, lfloat* Sb, lfloat* Tb,
    gfloat* inw, gfloat* inb, gfloat* ow, gfloat* ob,
    gfloat* w1,  gfloat* b1,  gfloat* w2, gfloat* b2,
    gfloat* g1,  gfloat* be1, gfloat* g2, gfloat* be2) {
  const int tid  = threadIdx.x;
  const int wave = tid >> 5;
  const int lane = tid & 31;
  const int r    = lane & 15;
  const int rh   = (lane >> 4) * 8;

  // 1) QKV = IN @ inw^T + inb   (64 x 192) -> Q | K | V, 48 tiles, 3 per wave
  for (int t = wave; t < 48; t += NWV) {
    int tm = t / 12, tc = t % 12;
    v8f acc = {};
    acc = tile_gemm_g<true>(IN + tm * 16 * ST, ST, inw + tc * 16 * ND, ND, ND, acc);
    float bias = inb[tc * 16 + r];
    lfloat* dst = (tc < 4 ? Q : (tc < 8 ? Kb : V));
    int cn = (tc & 3) * 16;
#pragma unroll
    for (int i = 0; i < 8; ++i)
      dst[(tm * 16 + rh + i) * ST + cn + r] = acc[i] + bias;
  }
  __syncthreads();

  // 2) scores = Q @ K^T * (1/sqrt(64))   -> Sb
  {
    int tm = wave >> 2, tn = wave & 3;
    v8f acc = {};
    acc = tile_gemm_l<true>(Q + tm * 16 * ST, ST, Kb + tn * 16 * ST, ST, ND, acc);
#pragma unroll
    for (int i = 0; i < 8; ++i)
      Sb[(tm * 16 + rh + i) * ST + tn * 16 + r] = acc[i] * 0.125f;
  }
  __syncthreads();

  // 3) row softmax (64 rows, 8 threads/row)
  {
    int row = tid >> 3, sub = tid & 7;
    lfloat* rp = Sb + row * ST;
    float m = -3.4e38f;
    for (int j = sub; j < 64; j += 8) m = fmaxf(m, rp[j]);
    for (int o = 1; o < 8; o <<= 1) m = fmaxf(m, __shfl_xor(m, o));
    float s = 0.f;
    for (int j = sub; j < 64; j += 8) { float e = __expf(rp[j] - m); rp[j] = e; s += e; }
    for (int o = 1; o < 8; o <<= 1) s += __shfl_xor(s, o);
    float inv = 1.f / s;
    for (int j = sub; j < 64; j += 8) rp[j] *= inv;
  }
  __syncthreads();

  // 4) att = softmax @ V  -> Tb
  {
    int tm = wave >> 2, tn = wave & 3;
    v8f acc = {};
    acc = tile_gemm_l<false>(Sb + tm * 16 * ST, ST, V + tn * 16, ST, ND, acc);
#pragma unroll
    for (int i = 0; i < 8; ++i)
      Tb[(tm * 16 + rh + i) * ST + tn * 16 + r] = acc[i];
  }
  __syncthreads();

  // 5) proj: att @ ow^T + ob + IN  -> Sb (pre-LN1)
  {
    int tm = wave >> 2, tn = wave & 3;
    v8f acc = {};
    acc = tile_gemm_g<true>(Tb + tm * 16 * ST, ST, ow + tn * 16 * ND, ND, ND, acc);
    float bias = ob[tn * 16 + r];
#pragma unroll
    for (int i = 0; i < 8; ++i) {
      int row = tm * 16 + rh + i, col = tn * 16 + r;
      Sb[row * ST + col] = acc[i] + bias + IN[row * ST + col];
    }
  }
  __syncthreads();

  // 6) LN1 -> Q (y1)
  ln_rows(Sb, Q, g1, be1);
  __syncthreads();

  // 7) ff1 = relu(y1 @ w1^T + b1) -> Tb
  {
    int tm = wave >> 2, tn = wave & 3;
    v8f acc = {};
    acc = tile_gemm_g<true>(Q + tm * 16 * ST, ST, w1 + tn * 16 * ND, ND, ND, acc);
    float bias = b1[tn * 16 + r];
#pragma unroll
    for (int i = 0; i < 8; ++i)
      Tb[(tm * 16 + rh + i) * ST + tn * 16 + r] = fmaxf(acc[i] + bias, 0.f);
  }
  __syncthreads();

  // 8) ff2 = Tb @ w2^T + b2 + y1 -> Sb (pre-LN2)
  {
    int tm = wave >> 2, tn = wave & 3;
    v8f acc = {};
    acc = tile_gemm_g<true>(Tb + tm * 16 * ST, ST, w2 + tn * 16 * ND, ND, ND, acc);
    float bias = b2[tn * 16 + r];
#pragma unroll
    for (int i = 0; i < 8; ++i) {
      int row = tm * 16 + rh + i, col = tn * 16 + r;
      Sb[row * ST + col] = acc[i] + bias + Q[row * ST + col];
    }
  }
  __syncthreads();

  // 9) LN2 -> OUT
  ln_rows(Sb, OUT, g2, be2);
  __syncthreads();
}

__global__ __launch_bounds__(NTH)
void gcrnn_kernel(const float* __restrict__ xp,   const float* __restrict__ adjp,
                  const float* __restrict__ h0p,
                  const float* __restrict__ inwp, const float* __restrict__ inbp,
                  const float* __restrict__ owp,  const float* __restrict__ obp,
                  const float* __restrict__ w1p,  const float* __restrict__ b1p,
                  const float* __restrict__ w2p,  const float* __restrict__ b2p,
                  const float* __restrict__ g1p,  const float* __restrict__ be1p,
                  const float* __restrict__ g2p,  const float* __restrict__ be2p,
                  const float* __restrict__ uwp,  const float* __restrict__ ubp,
                  const float* __restrict__ cwp,  const float* __restrict__ cbp,
                  float* __restrict__ outp) {
  extern __shared__ float smraw[];
  lfloat* sm   = (lfloat*)smraw;           // one-time generic->LDS cast
  gfloat* x    = (gfloat*)xp;    gfloat* adj  = (gfloat*)adjp;
  gfloat* h0   = (gfloat*)h0p;
  gfloat* inw  = (gfloat*)inwp;  gfloat* inb  = (gfloat*)inbp;
  gfloat* ow   = (gfloat*)owp;   gfloat* ob   = (gfloat*)obp;
  gfloat* w1   = (gfloat*)w1p;   gfloat* b1   = (gfloat*)b1p;
  gfloat* w2   = (gfloat*)w2p;   gfloat* b2   = (gfloat*)b2p;
  gfloat* g1   = (gfloat*)g1p;   gfloat* be1  = (gfloat*)be1p;
  gfloat* g2   = (gfloat*)g2p;   gfloat* be2  = (gfloat*)be2p;
  gfloat* uw   = (gfloat*)uwp;   gfloat* ub   = (gfloat*)ubp;
  gfloat* cw   = (gfloat*)cwp;   gfloat* cb   = (gfloat*)cbp;
  gmfloat* out = (gmfloat*)outp;

  lfloat* HY  = sm;                 // h in y-layout (n rows, d cols), persistent
  lfloat* YB  = sm + 1 * 64 * ST;   // encoder layer-0 output
  lfloat* EB  = sm + 2 * 64 * ST;   // encoder layer-1 output (eps^T)
  lfloat* SCR = sm + 3 * 64 * ST;   // 5 x 64 x ST scratch
  lfloat* Q  = SCR;
  lfloat* Kb = SCR + 1 * 64 * ST;
  lfloat* V  = SCR + 2 * 64 * ST;
  lfloat* Sb = SCR + 3 * 64 * ST;
  lfloat* Tb = SCR + 4 * 64 * ST;
  // diffusion-conv buffers alias encoder scratch (phases are barrier-separated)
  lfloat* XG = SCR;                 // 80 x ST (68 real channels + 12 zero pad rows)
  lfloat* X1 = SCR + 80 * ST;
  lfloat* X2 = SCR + 160 * ST;

  const int b    = blockIdx.x;
  const int tid  = threadIdx.x;
  const int wave = tid >> 5;
  const int lane = tid & 31;
  const int r    = lane & 15;
  const int rh   = (lane >> 4) * 8;
  const int ko   = (lane >> 4) << 1;

  // L2-warm the weights (reused 96x by all 128 workgroups) -> global_prefetch_b8
  for (int i = tid; i < (2 * 192 * ND) / 128; i += NTH) __builtin_prefetch(inwp + i * 128, 0, 3);
  for (int i = tid; i < (ND * 204 + 127) / 128; i += NTH) {
    __builtin_prefetch(uwp + i * 128, 0, 3);
    __builtin_prefetch(cwp + i * 128, 0, 3);
  }

  // init recurrent state: HY[n][d] = h0[b][d][n]
  for (int i = tid; i < 64 * 64; i += NTH) {
    int dd = i >> 6, n = i & 63;
    HY[n * ST + dd] = h0[(b * ND + dd) * NN + n];
  }
  __syncthreads();

  gfloat* times = x + (b * NF + 2) * NN * NS;   // x[b,2,0,:]

  for (int ts = 0; ts < NS; ++ts) {
    float dv = (ts == 0) ? 0.f : (times[ts] - times[ts - 1]);
    float s1 = sqrtf(fmaxf(1.f - dv, 0.f));
    float s2 = sqrtf(fmaxf(dv, 0.f));

    // ---- 2-layer transformer on y = h^T ----
    encoder_layer(HY, YB, Q, Kb, V, Sb, Tb,
                  inw,            inb,       ow,        ob,
                  w1,             b1,        w2,        b2,
                  g1,             be1,       g2,        be2);
    encoder_layer(YB, EB, Q, Kb, V, Sb, Tb,
                  inw + 192 * 64, inb + 192, ow + 4096, ob + 64,
                  w1 + 4096,      b1 + 64,   w2 + 4096, b2 + 64,
                  g1 + 64,        be1 + 64,  g2 + 64,   be2 + 64);

    // ---- build xg = [xin(4); hd(64); pad(12)]  (channels x nodes) ----
    for (int i = tid; i < 80 * 64; i += NTH) {
      int c = i >> 6, n = i & 63;
      float v;
      if (c < 3)        v = x[((b * NF + c) * NN + n) * NS + ts];
      else if (c == 3)  v = dv;
      else if (c < 68) { int dd = c - 4; v = s1 * HY[n * ST + dd] - s2 * EB[n * ST + dd]; }
      else              v = 0.f;
      XG[c * ST + n] = v;
    }
    __syncthreads();

    // ---- x1 = xg @ adj^T  (80x64, 20 tiles) ----
    for (int t = wave; t < 20; t += NWV) {
      int tm = t / 4, tn = t % 4;
      v8f acc = {};
      acc = tile_gemm_g<true>(XG + tm * 16 * ST, ST, adj + tn * 16 * NN, NN, NN, acc);
#pragma unroll
      for (int i = 0; i < 8; ++i)
        X1[(tm * 16 + rh + i) * ST + tn * 16 + r] = acc[i];
    }
    __syncthreads();

    // ---- x2 = x1 @ adj^T ----
    for (int t = wave; t < 20; t += NWV) {
      int tm = t / 4, tn = t % 4;
      v8f acc = {};
      acc = tile_gemm_g<true>(X1 + tm * 16 * ST, ST, adj + tn * 16 * NN, NN, NN, acc);
#pragma unroll
      for (int i = 0; i < 8; ++i)
        X2[(tm * 16 + rh + i) * ST + tn * 16 + r] = acc[i];
    }
    __syncthreads();

    // ---- gates: u = sigm(uw.z + ub), c = tanh(cw.z + cb); hn = u*hd + (1-u)*c ----
    {
      int tm = wave >> 2, tn = wave & 3;
      v8f au = {}, ac = {};
      gfloat* urow = uw + (tm * 16 + r) * 204;
      gfloat* crow = cw + (tm * 16 + r) * 204;
#pragma unroll
      for (int reg = 0; reg < 3; ++reg) {           // z = [xg | x1 | x2], 68 each
        lfloat* Z = (reg == 0) ? XG : ((reg == 1) ? X1 : X2);
        const int kbase = reg * 68;
#pragma unroll
        for (int kk = 0; kk < 68; kk += 4) {
          v2f bf, aU, aC;
          bf.x = Z[(kk + ko) * ST + tn * 16 + r];
          bf.y = Z[(kk + ko + 1) * ST + tn * 16 + r];
          aU.x = urow[kbase + kk + ko];
          aU.y = urow[kbase + kk + ko + 1];
          aC.x = crow[kbase + kk + ko];
          aC.y = crow[kbase + kk + ko + 1];
          au = wmma16(aU, bf, au);
          ac = wmma16(aC, bf, ac);
        }
      }
#pragma unroll
      for (int i = 0; i < 8; ++i) {
        int dd = tm * 16 + rh + i;                   // hidden dim (output channel)
        int n  = tn * 16 + r;                        // node
        float ug  = 1.f / (1.f + __expf(-(au[i] + ub[dd])));
        float cg  = tanhf(ac[i] + cb[dd]);
        float hdv = XG[(4 + dd) * ST + n];
        float hn  = ug * hdv + (1.f - ug) * cg;
        out[((b * NN + n) * ND + dd) * NS + ts] = hn;   // (B,N,D,S)
        HY[n * ST + dd] = hn;                            // next state (y-layout)
      }
    }
    __syncthreads();
  }
}

extern "C" void kernel_launch(void* const* d_in, const int* in_sizes, int n_in,
                              void* d_out, int out_size, void* d_ws, size_t ws_size,
                              hipStream_t stream) {
  (void)in_sizes; (void)n_in; (void)d_ws; (void)ws_size; (void)out_size;
  const float* x     = (const float*)d_in[0];
  const float* adj   = (const float*)d_in[1];
  /* d_in[2] lengths: unused by the reference forward pass */
  const float* h0    = (const float*)d_in[3];
  const float* t_inw = (const float*)d_in[4];
  const float* t_inb = (const float*)d_in[5];
  const float* t_ow  = (const float*)d_in[6];
  const float* t_ob  = (const float*)d_in[7];
  const float* t_w1  = (const float*)d_in[8];
  const float* t_b1  = (const float*)d_in[9];
  const float* t_w2  = (const float*)d_in[10];
  const float* t_b2  = (const float*)d_in[11];
  const float* t_g1  = (const float*)d_in[12];
  const float* t_be1 = (const float*)d_in[13];
  const float* t_g2  = (const float*)d_in[14];
  const float* t_be2 = (const float*)d_in[15];
  const float* uw    = (const float*)d_in[16];
  const float* ub    = (const float*)d_in[17];
  const float* cw    = (const float*)d_in[18];
  const float* cb    = (const float*)d_in[19];
  float* out = (float*)d_out;

  const size_t smem = (size_t)(3 * 64 * ST + 5 * 64 * ST) * sizeof(float); // 139264 B
  hipFuncSetAttribute((const void*)gcrnn_kernel,
                      hipFuncAttributeMaxDynamicSharedMemorySize, (int)smem);
  gcrnn_kernel<<<dim3(NB), dim3(NTH), smem, stream>>>(
      x, adj, h0, t_inw, t_inb, t_ow, t_ob, t_w1, t_b1, t_w2, t_b2,
      t_g1, t_be1, t_g2, t_be2, uw, ub, cw, cb, out);
}